// CrossNetwork_86844238725599
// MI455X (gfx1250) — compile-verified
//
#include <hip/hip_runtime.h>

// DCN cross network: xl_{l+1} = x0 * (xl . w_l) + b_l + xl
// B=32768, D=1024, 6 layers, all f32.
//
// Strategy (MI455X, wave32):
//  - one wave per batch row; lane holds 32 f32 of x0 and xl in VGPRs
//  - xl never touches memory between layers -> HBM traffic = read x0 + write out = 256 MB
//    (~11 us at 23.3 TB/s; compute ~1 GFLOP, far below the VALU roof -> memory bound)
//  - per-layer dot product: 32 per-lane FMAs, then cross-lane reduce with TWO chained
//    v_wmma_f32_16x16x4_f32 ops (no LDS, no shuffles, no dscnt stalls):
//      WMMA1: A=(p,0), B=ones       -> D1_r = t_r | t_{r+8}, t_m = p_m + p_{m+16}
//      u = sum_r D1_r               -> uA = sum(t_0..7) (lanes 0-15) | uB = sum(t_8..15)
//      WMMA2: A=(u,0), B=ones       -> D2[m,n] = uA + uB = s, replicated in EVERY lane
//  - w/b (24 KB each) reloaded per layer from cache (WGP$/L2 resident)
//  - non-temporal load/store for the 128 MB streams

typedef float v2f __attribute__((ext_vector_type(2)));
typedef float v4f __attribute__((ext_vector_type(4)));
typedef float v8f __attribute__((ext_vector_type(8)));

#define NUM_LAYERS 6
#define DIM 1024
#define CHUNKS 8            // 8 x float4 = 32 f32 per lane
#define WAVES_PER_BLOCK 8   // 256 threads/block = 8 rows/block

__global__ __launch_bounds__(256) void dcn_cross_kernel(
    const float* __restrict__ x0,
    const float* __restrict__ w,
    const float* __restrict__ b,
    float* __restrict__ out)
{
    const int lane = threadIdx.x & 31;
    const int wid  = threadIdx.x >> 5;
    const size_t row = (size_t)blockIdx.x * WAVES_PER_BLOCK + wid;

    const v4f* __restrict__ x0row  = reinterpret_cast<const v4f*>(x0 + row * DIM);
    v4f* __restrict__       outrow = reinterpret_cast<v4f*>(out + row * DIM);

    // lane L, chunk i covers columns [i*128 + L*4, i*128 + L*4 + 4)
    // -> each vector load instruction touches 512 contiguous bytes (perfect coalescing)
    v4f x0v[CHUNKS], xlv[CHUNKS];
#pragma unroll
    for (int i = 0; i < CHUNKS; ++i) {
        x0v[i] = __builtin_nontemporal_load(&x0row[i * 32 + lane]);
        xlv[i] = x0v[i];
    }

    // B operand for both reduction WMMAs: all ones (B[k,n] = 1)
    v2f bo; bo.x = 1.0f; bo.y = 1.0f;

#pragma unroll
    for (int l = 0; l < NUM_LAYERS; ++l) {
        const v4f* __restrict__ wrow = reinterpret_cast<const v4f*>(w + l * DIM);
        const v4f* __restrict__ brow = reinterpret_cast<const v4f*>(b + l * DIM);

        // per-lane partial of s = xl . w  (32 FMAs)
        float p = 0.0f;
#pragma unroll
        for (int i = 0; i < CHUNKS; ++i) {
            v4f wv = wrow[i * 32 + lane];
            p = fmaf(xlv[i].x, wv.x, p);
            p = fmaf(xlv[i].y, wv.y, p);
            p = fmaf(xlv[i].z, wv.z, p);
            p = fmaf(xlv[i].w, wv.w, p);
        }

        // ---- stage 1: fold lane halves.  A layout (16x4 f32): lane L supplies
        // A[M=L%16, K=2*(L/16)] in VGPR0 and K=2*(L/16)+1 in VGPR1.
        //   A=(p,0), B=ones  ->  D1[m,n] = p_m + p_{m+16} = t_m  (any n)
        // D1 VGPR r, lane L = t_{r + 8*(L>=16)}.
        v2f a1; a1.x = p; a1.y = 0.0f;
        v8f c0 = {};
        v8f d1 = __builtin_amdgcn_wmma_f32_16x16x4_f32(
            false, a1, false, bo, (short)0, c0, false, false);

        // sum the 8 D1 registers -> uA = sum(t_0..t_7) on lanes 0-15,
        //                           uB = sum(t_8..t_15) on lanes 16-31
        float u = ((d1[0] + d1[1]) + (d1[2] + d1[3])) +
                  ((d1[4] + d1[5]) + (d1[6] + d1[7]));

        // ---- stage 2: fold across halves with a second WMMA (no shuffle).
        //   A2[m,0] = uA (all m, from lanes 0-15), A2[m,2] = uB (lanes 16-31)
        //   D2[m,n] = uA + uB = s  -> replicated in every lane of every D VGPR
        v2f a2; a2.x = u; a2.y = 0.0f;
        v8f d2 = __builtin_amdgcn_wmma_f32_16x16x4_f32(
            false, a2, false, bo, (short)0, c0, false, false);
        const float s = d2[0];

        // xl = x0 * s + b + xl   (rank-1 update, elementwise)
#pragma unroll
        for (int i = 0; i < CHUNKS; ++i) {
            v4f bv = brow[i * 32 + lane];
            xlv[i].x = fmaf(x0v[i].x, s, xlv[i].x + bv.x);
            xlv[i].y = fmaf(x0v[i].y, s, xlv[i].y + bv.y);
            xlv[i].z = fmaf(x0v[i].z, s, xlv[i].z + bv.z);
            xlv[i].w = fmaf(x0v[i].w, s, xlv[i].w + bv.w);
        }
    }

#pragma unroll
    for (int i = 0; i < CHUNKS; ++i) {
        __builtin_nontemporal_store(xlv[i], &outrow[i * 32 + lane]);
    }
}

extern "C" void kernel_launch(void* const* d_in, const int* in_sizes, int n_in,
                              void* d_out, int out_size, void* d_ws, size_t ws_size,
                              hipStream_t stream) {
    (void)n_in; (void)d_ws; (void)ws_size; (void)out_size;
    const float* x0 = (const float*)d_in[0];
    const float* w  = (const float*)d_in[1];
    const float* b  = (const float*)d_in[2];
    float* out      = (float*)d_out;

    const int rows   = in_sizes[0] / DIM;          // 32768
    const int blocks = rows / WAVES_PER_BLOCK;     // 4096 full blocks (rows % 8 == 0)
    dcn_cross_kernel<<<blocks, 32 * WAVES_PER_BLOCK, 0, stream>>>(x0, w, b, out);
}